// PointerRubricBiEncoder_44246753084250
// MI455X (gfx1250) — compile-verified
//
#include <hip/hip_runtime.h>
#include <hip/hip_bf16.h>
#include <math.h>

// Problem constants (from reference setup_inputs)
#define B_SZ 32
#define R_SZ 64
#define L_SZ 4096
#define H_SZ 768

typedef __attribute__((ext_vector_type(2))) float v2f;
typedef __attribute__((ext_vector_type(8))) float v8f;

// ---------------------------------------------------------------------------
// Kernel 1: V[b,h] = sum_k W[h,k] * ans[b,k]   (i.e. V = ans @ W^T)
// GEMM M=32 (batch), N=768 (h), K=768 via V_WMMA_F32_16X16X4_F32.
// One wave (32 threads) per 16x16 output tile; EXEC is all-ones as required.
// A(16x4): lane<16 -> M=lane, VGPR v holds K = 2*half + v.
// B(4x16): VGPR v holds row K = v + 2*half, col N = lane&15; B[k][n]=W[n][k].
// D(16x16): VGPR g -> row M = g + 8*half, col N = lane&15.
// ---------------------------------------------------------------------------
__global__ __launch_bounds__(32) void prb_bilinear_wmma(
    const float* __restrict__ ans,   // (B,H)
    const float* __restrict__ W,     // (H,H) row-major: W[h*H + k]
    float* __restrict__ V)           // (B,H)
{
    const int nt   = blockIdx.x;          // 0..H/16-1
    const int mt   = blockIdx.y;          // 0..B/16-1
    const int lane = threadIdx.x;         // 0..31
    const int half = lane >> 4;
    const int l16  = lane & 15;

    const float* aptr = ans + (mt * 16 + l16) * H_SZ;  // A row = batch row
    const float* bptr = W   + (nt * 16 + l16) * H_SZ;  // B col n -> W row n

    v8f c = {};
#pragma unroll 8
    for (int k = 0; k < H_SZ; k += 4) {
        v2f a, bb;
        a[0]  = aptr[k + 2 * half + 0];
        a[1]  = aptr[k + 2 * half + 1];
        bb[0] = bptr[k + 2 * half + 0];
        bb[1] = bptr[k + 2 * half + 1];
        // (neg_a, A, neg_b, B, c_mod, C, reuse_a, reuse_b)
        c = __builtin_amdgcn_wmma_f32_16x16x4_f32(
                false, a, false, bb, (short)0, c, false, false);
    }

    float* out = V + (mt * 16 + 8 * half) * H_SZ + nt * 16 + l16;
#pragma unroll
    for (int g = 0; g < 8; ++g) out[g * H_SZ] = c[g];
}

// ---------------------------------------------------------------------------
// Kernel 2: t[b,l] = dot(hidden[b,l,:], V[b,:])
// The ONLY pass over the 402MB rubric_hidden tensor -> bandwidth bound
// (~17us floor at 23.3 TB/s). One wave per row l; lanes read float4 with
// stride 32 -> fully coalesced global_load_b128 (512B per wave per issue).
// V[b,:] staged into LDS via CDNA5 async-to-LDS (GLOBAL_LOAD_ASYNC_TO_LDS_
// B128 + s_wait_asynccnt); vsh is the only LDS object so its offset is 0.
// ---------------------------------------------------------------------------
__global__ __launch_bounds__(256) void prb_row_dot(
    const float* __restrict__ hid,   // (B,L,H)
    const float* __restrict__ V,     // (B,H)
    float* __restrict__ t)           // (B,L)
{
    const int b    = blockIdx.y;
    const int w    = threadIdx.x >> 5;
    const int lane = threadIdx.x & 31;
    const int l    = blockIdx.x * 8 + w;

    __shared__ __align__(16) float vsh[H_SZ];

    // Async-stage V[b,:] (3KB) straight into LDS: 192 x b128 transfers.
    if (threadIdx.x < H_SZ / 4) {
        unsigned     ldsoff = threadIdx.x * 16u;            // byte offset in LDS
        const float* gp     = V + b * H_SZ + threadIdx.x * 4;
        asm volatile("global_load_async_to_lds_b128 %0, %1, off"
                     :: "v"(ldsoff), "v"(gp) : "memory");
    }
    asm volatile("s_wait_asynccnt 0" ::: "memory");
    __syncthreads();

    const float*  row  = hid + ((size_t)b * L_SZ + l) * H_SZ;
    const float4* row4 = (const float4*)row;
    const float4* vsh4 = (const float4*)vsh;
    __builtin_prefetch(row + 8 * H_SZ, 0, 1);   // global_prefetch_b8

    float s = 0.f;
#pragma unroll
    for (int j = 0; j < H_SZ / 128; ++j) {      // 6 iterations
        float4 r = row4[lane + 32 * j];
        float4 v = vsh4[lane + 32 * j];
        s += r.x * v.x + r.y * v.y + r.z * v.z + r.w * v.w;
    }

    for (int off = 16; off > 0; off >>= 1)
        s += __shfl_down(s, off, 32);

    if (lane == 0) t[b * L_SZ + l] = s;
}

// ---------------------------------------------------------------------------
// Kernel 3: per-(b,r) span-mean of t, mask -> -inf, softmax over R.
// One block (8 waves) per batch; wave w handles rubrics w*8..w*8+7.
// ---------------------------------------------------------------------------
__global__ __launch_bounds__(256) void prb_score_softmax(
    const float* __restrict__ t,           // (B,L)
    const int* __restrict__ span,          // (B,R,2)
    const unsigned char* __restrict__ msk, // (B,R) bool
    const float* __restrict__ bias,        // (1,)
    float* __restrict__ out)               // (B,R)
{
    const int b    = blockIdx.x;
    const int w    = threadIdx.x >> 5;
    const int lane = threadIdx.x & 31;

    __shared__ float sc[R_SZ];
    const float bs = bias[0];

    for (int i = 0; i < 8; ++i) {
        const int r  = w * 8 + i;
        const int s0 = span[(b * R_SZ + r) * 2 + 0];
        const int e0 = span[(b * R_SZ + r) * 2 + 1];
        float acc = 0.f;
        for (int l = s0 + lane; l < e0; l += 32) acc += t[b * L_SZ + l];
        for (int off = 16; off > 0; off >>= 1)
            acc += __shfl_down(acc, off, 32);
        if (lane == 0) {
            int len = e0 - s0; if (len < 1) len = 1;
            float v = acc / (float)len + bs;
            sc[r] = msk[b * R_SZ + r] ? v : -__builtin_inff();
        }
    }
    __syncthreads();

    if (threadIdx.x < 32) {
        float v0 = sc[lane], v1 = sc[lane + 32];
        float m = fmaxf(v0, v1);
        for (int off = 16; off > 0; off >>= 1)
            m = fmaxf(m, __shfl_xor(m, off, 32));
        float e0 = expf(v0 - m), e1 = expf(v1 - m);
        float sm = e0 + e1;
        for (int off = 16; off > 0; off >>= 1)
            sm += __shfl_xor(sm, off, 32);
        out[b * R_SZ + lane]      = e0 / sm;
        out[b * R_SZ + lane + 32] = e1 / sm;
    }
}

extern "C" void kernel_launch(void* const* d_in, const int* in_sizes, int n_in,
                              void* d_out, int out_size, void* d_ws, size_t ws_size,
                              hipStream_t stream) {
    const float*         ans  = (const float*)d_in[0];         // (B,H)
    const float*         hid  = (const float*)d_in[1];         // (B,L,H)
    const int*           span = (const int*)d_in[2];           // (B,R,2)
    const unsigned char* msk  = (const unsigned char*)d_in[3]; // (B,R) bool
    const float*         W    = (const float*)d_in[4];         // (1,H,H)
    const float*         bias = (const float*)d_in[5];         // (1,)
    float*               out  = (float*)d_out;                 // (B,R)

    float* V = (float*)d_ws;            // B*H floats
    float* t = V + B_SZ * H_SZ;         // B*L floats

    prb_bilinear_wmma<<<dim3(H_SZ / 16, B_SZ / 16), 32, 0, stream>>>(ans, W, V);
    prb_row_dot<<<dim3(L_SZ / 8, B_SZ), 256, 0, stream>>>(hid, V, t);
    prb_score_softmax<<<B_SZ, 256, 0, stream>>>(t, span, msk, bias, out);
}